// Mamba_37374805409896
// MI455X (gfx1250) — compile-verified
//
#include <hip/hip_runtime.h>
#include <hip/hip_bf16.h>

#define D_MODEL 768
#define N_LAYER 4
#define VOCAB   50280
#define D_STATE 16
#define D_CONV  4
#define D_INNER 1536
#define DT_RANK 48
#define SEQ     1024
#define EPS     1e-5f

typedef __attribute__((ext_vector_type(2))) float v2f;
typedef __attribute__((ext_vector_type(8))) float v8f;

// ---------------------------------------------------------------------------
// Embedding gather: h[l][d] = emb[ids[l]][d]
// ---------------------------------------------------------------------------
__global__ void embed_kernel(const int* __restrict__ ids,
                             const float* __restrict__ emb,
                             float* __restrict__ h) {
    const int l = blockIdx.x;
    const int row = ids[l];
    for (int d = threadIdx.x; d < D_MODEL; d += blockDim.x)
        h[(size_t)l * D_MODEL + d] = emb[(size_t)row * D_MODEL + d];
}

// ---------------------------------------------------------------------------
// LayerNorm over last dim (D = 768), one block per row
// ---------------------------------------------------------------------------
__global__ void layernorm_kernel(const float* __restrict__ x,
                                 const float* __restrict__ w,
                                 const float* __restrict__ b,
                                 float* __restrict__ out, int D) {
    __shared__ float s1[256], s2[256];
    const int row = blockIdx.x;
    const int tid = threadIdx.x;
    float a = 0.f, q = 0.f;
    for (int d = tid; d < D; d += blockDim.x) {
        float v = x[(size_t)row * D + d];
        a += v; q += v * v;
    }
    s1[tid] = a; s2[tid] = q;
    __syncthreads();
    for (int s = 128; s > 0; s >>= 1) {
        if (tid < s) { s1[tid] += s1[tid + s]; s2[tid] += s2[tid + s]; }
        __syncthreads();
    }
    const float mean = s1[0] / (float)D;
    const float var  = s2[0] / (float)D - mean * mean;
    const float rstd = rsqrtf(var + EPS);
    for (int d = tid; d < D; d += blockDim.x) {
        float v = x[(size_t)row * D + d];
        out[(size_t)row * D + d] = (v - mean) * rstd * w[d] + b[d];
    }
}

// ---------------------------------------------------------------------------
// WMMA f32 GEMM:  O[M,N] = X[M,K] * W[N,K]^T  (+bias, +softplus, +add)
// One wave per 64x64 tile, 16 v8f accumulators, K pipelined with two
// register fragment buffers (loads for step s+2 overlap wmmas of step s).
// A frag (16x4 f32): lane L holds row m0+L%16, K pair at 2*(L/16).
// B frag (4x16 f32): lane L holds col n0+L%16, K pair at 2*(L/16).
// C/D (16x16 f32):   vgpr r -> row r + 8*(L/16), col L%16.
// Requires M % 64 == 0, K % 8 == 0 (true for every call); N arbitrary
// (B rows clamped, stores masked).
// ---------------------------------------------------------------------------
template<bool HAS_BIAS, bool HAS_ADD, bool SOFTPLUS>
__global__ __launch_bounds__(32)
void gemm_wmma_f32(const float* __restrict__ X, const float* __restrict__ W,
                   const float* __restrict__ bias, const float* __restrict__ add,
                   float* __restrict__ O,
                   int N, int K, int ldx, int ldw, int ldo) {
    const int lane = threadIdx.x;
    const int lm = lane & 15;
    const int lh = lane >> 4;
    const int m0 = blockIdx.x * 64;
    const int n0 = blockIdx.y * 64;

    const float* xptr[4];
    const float* wptr[4];
    #pragma unroll
    for (int i = 0; i < 4; ++i)
        xptr[i] = X + (size_t)(m0 + i * 16 + lm) * ldx + 2 * lh;
    #pragma unroll
    for (int j = 0; j < 4; ++j) {
        int nr = n0 + j * 16 + lm;
        if (nr > N - 1) nr = N - 1;          // clamp: loads always in-bounds
        wptr[j] = W + (size_t)nr * ldw + 2 * lh;
    }

    v8f acc[4][4];
    #pragma unroll
    for (int i = 0; i < 4; ++i)
        #pragma unroll
        for (int j = 0; j < 4; ++j)
            acc[i][j] = (v8f)0.f;

    auto mma16 = [&](const v2f (&A)[4], const v2f (&B)[4]) {
        #pragma unroll
        for (int j = 0; j < 4; ++j)
            #pragma unroll
            for (int i = 0; i < 4; ++i)
                acc[i][j] = __builtin_amdgcn_wmma_f32_16x16x4_f32(
                    false, A[i], false, B[j], (short)0, acc[i][j], false, false);
    };

    // Prologue: fill both fragment buffers (K >= 8 guaranteed).
    v2f a0[4], b0[4], a1[4], b1[4];
    #pragma unroll
    for (int i = 0; i < 4; ++i) a0[i] = *(const v2f*)(xptr[i]);
    #pragma unroll
    for (int j = 0; j < 4; ++j) b0[j] = *(const v2f*)(wptr[j]);
    #pragma unroll
    for (int i = 0; i < 4; ++i) a1[i] = *(const v2f*)(xptr[i] + 4);
    #pragma unroll
    for (int j = 0; j < 4; ++j) b1[j] = *(const v2f*)(wptr[j] + 4);

    const int steps = K >> 2;                 // even for all call sites
    #pragma unroll 1
    for (int s = 0; s < steps - 2; s += 2) {
        const int k0 = (s + 2) << 2;
        const int k1 = (s + 3) << 2;
        mma16(a0, b0);
        #pragma unroll
        for (int i = 0; i < 4; ++i) a0[i] = *(const v2f*)(xptr[i] + k0);
        #pragma unroll
        for (int j = 0; j < 4; ++j) b0[j] = *(const v2f*)(wptr[j] + k0);
        mma16(a1, b1);
        #pragma unroll
        for (int i = 0; i < 4; ++i) a1[i] = *(const v2f*)(xptr[i] + k1);
        #pragma unroll
        for (int j = 0; j < 4; ++j) b1[j] = *(const v2f*)(wptr[j] + k1);
    }
    mma16(a0, b0);
    mma16(a1, b1);

    // Epilogue: masked stores (covers partial N tiles).
    #pragma unroll
    for (int j = 0; j < 4; ++j) {
        const int col = n0 + j * 16 + lm;
        if (col >= N) continue;               // store-side masking only
        const float bv = HAS_BIAS ? bias[col] : 0.f;
        #pragma unroll
        for (int i = 0; i < 4; ++i) {
            #pragma unroll
            for (int r = 0; r < 8; ++r) {
                const int row = m0 + i * 16 + 8 * lh + r;
                float v = acc[i][j][r] + bv;
                if (SOFTPLUS) v = (v > 20.f) ? v : log1pf(__expf(v));
                if (HAS_ADD)  v += add[(size_t)row * ldo + col];
                O[(size_t)row * ldo + col] = v;
            }
        }
    }
}

// ---------------------------------------------------------------------------
// Causal depthwise conv (D_CONV=4) + bias + SiLU.
// Input: x part of xz ([L][3072], cols 0..1535). Output xc [L][1536].
// ---------------------------------------------------------------------------
__global__ void conv_silu_kernel(const float* __restrict__ xz,
                                 const float* __restrict__ cw,
                                 const float* __restrict__ cb,
                                 float* __restrict__ xc) {
    const int l = blockIdx.x;
    const int c = blockIdx.y * blockDim.x + threadIdx.x;
    float acc = cb[c];
    #pragma unroll
    for (int j = 0; j < D_CONV; ++j) {
        int ll = l - (D_CONV - 1) + j;
        if (ll >= 0)
            acc += cw[c * D_CONV + j] * xz[(size_t)ll * (2 * D_INNER) + c];
    }
    xc[(size_t)l * D_INNER + c] = acc / (1.f + __expf(-acc));
}

// ---------------------------------------------------------------------------
// Selective scan. One lane per (channel d, state n): 1536*16 = 24576 lanes.
// State kept in a register; einsum over n via shfl_xor within 16-lane groups.
// Fuses +x*D and the silu(z) gate into the y write.
// ---------------------------------------------------------------------------
__global__ void scan_kernel(const float* __restrict__ xc,   // [L][1536] conv+silu
                            const float* __restrict__ dt,   // [L][1536]
                            const float* __restrict__ xdbl, // [L][80] (B at 48, C at 64)
                            const float* __restrict__ A_log,// [1536][16]
                            const float* __restrict__ Dp,   // [1536]
                            const float* __restrict__ xz,   // [L][3072] (z at 1536)
                            float* __restrict__ y) {        // [L][1536]
    const int t = blockIdx.x * blockDim.x + threadIdx.x;    // 0..24575
    const int n = t & 15;
    const int d = t >> 4;
    const float Ac = -__expf(A_log[d * D_STATE + n]);
    const float Dv = Dp[d];
    float hstate = 0.f;
    for (int l = 0; l < SEQ; ++l) {
        const float dtv = dt[(size_t)l * D_INNER + d];
        const float xv  = xc[(size_t)l * D_INNER + d];
        const float Bv  = xdbl[(size_t)l * 80 + DT_RANK + n];
        const float Cv  = xdbl[(size_t)l * 80 + DT_RANK + D_STATE + n];
        hstate = __expf(dtv * Ac) * hstate + (dtv * xv) * Bv;
        float contrib = hstate * Cv;
        contrib += __shfl_xor(contrib, 1, 32);
        contrib += __shfl_xor(contrib, 2, 32);
        contrib += __shfl_xor(contrib, 4, 32);
        contrib += __shfl_xor(contrib, 8, 32);
        if (n == 0) {
            const float zv = xz[(size_t)l * (2 * D_INNER) + D_INNER + d];
            y[(size_t)l * D_INNER + d] =
                (contrib + xv * Dv) * (zv / (1.f + __expf(-zv)));
        }
    }
}

// ---------------------------------------------------------------------------
// Host-side orchestration
// ---------------------------------------------------------------------------
static inline dim3 gemm_grid(int M, int N) {
    return dim3(M / 64, (N + 63) / 64);
}

extern "C" void kernel_launch(void* const* d_in, const int* in_sizes, int n_in,
                              void* d_out, int out_size, void* d_ws, size_t ws_size,
                              hipStream_t stream) {
    (void)in_sizes; (void)n_in; (void)out_size; (void)ws_size;
    const int*   ids    = (const int*)d_in[0];
    const float* emb    = (const float*)d_in[1];
    const float* ln_w   = (const float*)d_in[2];
    const float* ln_b   = (const float*)d_in[3];
    const float* inp_w  = (const float*)d_in[4];
    const float* conv_w = (const float*)d_in[5];
    const float* conv_b = (const float*)d_in[6];
    const float* xp_w   = (const float*)d_in[7];
    const float* dtp_w  = (const float*)d_in[8];
    const float* dtp_b  = (const float*)d_in[9];
    const float* A_log  = (const float*)d_in[10];
    const float* Dp     = (const float*)d_in[11];
    const float* outp_w = (const float*)d_in[12];
    const float* lnf_w  = (const float*)d_in[13];
    const float* lnf_b  = (const float*)d_in[14];
    float* logits = (float*)d_out;

    float* ws = (float*)d_ws;
    float* h    = ws;                      ws += (size_t)SEQ * D_MODEL;
    float* hn   = ws;                      ws += (size_t)SEQ * D_MODEL;
    float* xz   = ws;                      ws += (size_t)SEQ * 2 * D_INNER;
    float* xc   = ws;                      ws += (size_t)SEQ * D_INNER;
    float* xdbl = ws;                      ws += (size_t)SEQ * 80;
    float* dt   = ws;                      ws += (size_t)SEQ * D_INNER;
    float* y    = ws;                      ws += (size_t)SEQ * D_INNER;

    embed_kernel<<<SEQ, 256, 0, stream>>>(ids, emb, h);

    for (int i = 0; i < N_LAYER; ++i) {
        layernorm_kernel<<<SEQ, 256, 0, stream>>>(
            h, ln_w + (size_t)i * D_MODEL, ln_b + (size_t)i * D_MODEL, hn, D_MODEL);

        // xz = hn @ in_proj_w^T   [1024 x 3072]
        gemm_wmma_f32<false, false, false><<<gemm_grid(SEQ, 2 * D_INNER), 32, 0, stream>>>(
            hn, inp_w + (size_t)i * 2 * D_INNER * D_MODEL, nullptr, nullptr, xz,
            2 * D_INNER, D_MODEL, D_MODEL, D_MODEL, 2 * D_INNER);

        // depthwise causal conv + bias + silu -> xc
        conv_silu_kernel<<<dim3(SEQ, D_INNER / 256), 256, 0, stream>>>(
            xz, conv_w + (size_t)i * D_INNER * D_CONV, conv_b + (size_t)i * D_INNER, xc);

        // x_dbl = xc @ x_proj_w^T   [1024 x 80]
        gemm_wmma_f32<false, false, false><<<gemm_grid(SEQ, DT_RANK + 2 * D_STATE), 32, 0, stream>>>(
            xc, xp_w + (size_t)i * (DT_RANK + 2 * D_STATE) * D_INNER, nullptr, nullptr, xdbl,
            DT_RANK + 2 * D_STATE, D_INNER, D_INNER, D_INNER, 80);

        // dt = softplus(x_dbl[:, :48] @ dt_proj_w^T + dt_proj_b)   [1024 x 1536]
        gemm_wmma_f32<true, false, true><<<gemm_grid(SEQ, D_INNER), 32, 0, stream>>>(
            xdbl, dtp_w + (size_t)i * D_INNER * DT_RANK,
            dtp_b + (size_t)i * D_INNER, nullptr, dt,
            D_INNER, DT_RANK, 80, DT_RANK, D_INNER);

        // selective scan (+ x*D, * silu(z)) -> y
        scan_kernel<<<(D_INNER * D_STATE) / 256, 256, 0, stream>>>(
            xc, dt, xdbl, A_log + (size_t)i * D_INNER * D_STATE,
            Dp + (size_t)i * D_INNER, xz, y);

        // h = y @ out_proj_w^T + h (residual, in-place safe per-element)
        gemm_wmma_f32<false, true, false><<<gemm_grid(SEQ, D_MODEL), 32, 0, stream>>>(
            y, outp_w + (size_t)i * D_MODEL * D_INNER, nullptr, h, h,
            D_MODEL, D_INNER, D_INNER, D_INNER, D_MODEL);
    }

    layernorm_kernel<<<SEQ, 256, 0, stream>>>(h, lnf_w, lnf_b, hn, D_MODEL);

    // logits = hn @ emb^T   [1024 x 50280]
    gemm_wmma_f32<false, false, false><<<gemm_grid(SEQ, VOCAB), 32, 0, stream>>>(
        hn, emb, nullptr, nullptr, logits,
        VOCAB, D_MODEL, D_MODEL, D_MODEL, VOCAB);
}